// LowPassFilterAugmentation_88287347736691
// MI455X (gfx1250) — compile-verified
//
#include <hip/hip_runtime.h>
#include <math.h>

// ---------------------------------------------------------------------------
// Biquad low-pass over (64, 524288) f32, via chunked parallel linear-recurrence
// scan. Inner recurrence advanced 16 samples/step with V_WMMA_F32_16X16X4_F32:
// each wave runs 16 independent chunks in lockstep, Y = T*X + H*S per step.
// Staging global<->LDS uses CDNA5 async-to/from-LDS ops when available.
// ---------------------------------------------------------------------------

typedef __attribute__((ext_vector_type(2))) float v2f;
typedef __attribute__((ext_vector_type(8))) float v8f;
typedef __attribute__((__vector_size__(4 * sizeof(int)))) int v4i; // async op payload type

#if defined(__has_builtin)
#if __has_builtin(__builtin_amdgcn_global_load_async_to_lds_b128) && \
    __has_builtin(__builtin_amdgcn_global_store_async_from_lds_b128) && \
    __has_builtin(__builtin_amdgcn_s_wait_asynccnt)
#define USE_ASYNC_LDS 1
#endif
#endif
#ifndef USE_ASYNC_LDS
#define USE_ASYNC_LDS 0
#endif

#define SRATE   16000.0f
#define TLEN    524288
#define NROW    64
#define CHUNK   2048
#define CPR     (TLEN / CHUNK)      // 256 chunks per row
#define NCHUNK  (NROW * CPR)        // 16384 chunks total
#define GROUPS  (CHUNK / 64)        // 32 LDS staging groups per chunk
#define CH      68                  // padded LDS row stride (floats, 16B aligned)

// workspace layout (float offsets); needs (1024 + 4*NCHUNK)*4 B ~= 267 KB
#define WS_T  0                     // 16x16 impulse-response matrix (row major)
#define WS_H  256                   // 16x4  [hy1 hy2 hx1 hx2]
#define WS_M2 320                   // 2x2   A^CHUNK
#define WS_P  1024                  // per-chunk zero-state finals (2 floats)
#define WS_S  (1024 + 2*NCHUNK)     // per-chunk start states (2 floats)

// ---------------------------------------------------------------------------
// Setup: single thread derives biquad coeffs from cutoff and builds all small
// constant tables in workspace.
// ---------------------------------------------------------------------------
__global__ void setup_kernel(const float* __restrict__ cutoff, float* __restrict__ ws) {
  if (threadIdx.x != 0 || blockIdx.x != 0) return;
  const float c  = cutoff[0];
  const float w0 = 2.0f * 3.14159265358979323846f * c / SRATE;
  const float sw = sinf(w0), cw = cosf(w0);
  const float alpha = sw * 0.70710678118654752440f;   // sin(w0)/(2Q), Q=1/sqrt2
  const float ia0 = 1.0f / (1.0f + alpha);
  const float b0 = 0.5f * (1.0f - cw) * ia0;
  const float b1 = (1.0f - cw) * ia0;
  const float b2 = b0;
  const float a1 = -2.0f * cw * ia0;
  const float a2 = (1.0f - alpha) * ia0;

  float f[16], hy1[16], hy2[16], hx1[16], hx2[16];
  { float y1=0.f, y2=0.f;                         // response to x[0]=1
    for (int n=0;n<16;n++){ float in = (n==0?b0:(n==1?b1:(n==2?b2:0.f)));
      float y = in - a1*y1 - a2*y2; f[n]=y; y2=y1; y1=y; } }
  { float y1=1.f, y2=0.f;                         // response to y[-1]=1
    for (int n=0;n<16;n++){ float y = -a1*y1 - a2*y2; hy1[n]=y; y2=y1; y1=y; } }
  { float y1=0.f, y2=1.f;                         // response to y[-2]=1
    for (int n=0;n<16;n++){ float y = -a1*y1 - a2*y2; hy2[n]=y; y2=y1; y1=y; } }
  { float y1=0.f, y2=0.f;                         // response to x[-1]=1
    for (int n=0;n<16;n++){ float in = (n==0?b1:(n==1?b2:0.f));
      float y = in - a1*y1 - a2*y2; hx1[n]=y; y2=y1; y1=y; } }
  { float y1=0.f, y2=0.f;                         // response to x[-2]=1
    for (int n=0;n<16;n++){ float in = (n==0?b2:0.f);
      float y = in - a1*y1 - a2*y2; hx2[n]=y; y2=y1; y1=y; } }

  for (int n=0;n<16;n++)
    for (int m=0;m<16;m++)
      ws[WS_T + n*16 + m] = (n>=m) ? f[n-m] : 0.0f;
  for (int n=0;n<16;n++) {
    ws[WS_H + n*4+0]=hy1[n]; ws[WS_H + n*4+1]=hy2[n];
    ws[WS_H + n*4+2]=hx1[n]; ws[WS_H + n*4+3]=hx2[n];
  }
  // A = [[-a1,-a2],[1,0]]; A^2048 by 11 squarings (maps (y[-1],y[-2]) -> (y[L-1],y[L-2]))
  float m00=-a1, m01=-a2, m10=1.0f, m11=0.0f;
  for (int it=0; it<11; ++it) {
    float t00=m00*m00+m01*m10, t01=m00*m01+m01*m11;
    float t10=m10*m00+m11*m10, t11=m10*m01+m11*m11;
    m00=t00; m01=t01; m10=t10; m11=t11;
  }
  ws[WS_M2+0]=m00; ws[WS_M2+1]=m01; ws[WS_M2+2]=m10; ws[WS_M2+3]=m11;
}

// ---------------------------------------------------------------------------
// Scan: per row, compose chunk summaries: sigma_{k+1} = A^L * sigma_k + p_k
// ---------------------------------------------------------------------------
__global__ void scan_kernel(float* __restrict__ ws) {
  const int r = threadIdx.x;
  if (r >= NROW) return;
  const float m00=ws[WS_M2+0], m01=ws[WS_M2+1], m10=ws[WS_M2+2], m11=ws[WS_M2+3];
  const float* p = ws + WS_P;
  float*       s = ws + WS_S;
  float s1=0.f, s2=0.f;
  #pragma unroll 8
  for (int k=0;k<CPR;k++) {
    const int i = r*CPR + k;
    s[2*i+0]=s1; s[2*i+1]=s2;
    const float p1=p[2*i+0], p2=p[2*i+1];
    const float n1 = fmaf(m00,s1,fmaf(m01,s2,p1));
    const float n2 = fmaf(m10,s1,fmaf(m11,s2,p2));
    s1=n1; s2=n2;
  }
}

// bring upper-half-lane value into lower half (and vice versa)
__device__ inline float swap_half(float v) {
  const int lane = (int)(threadIdx.x & 31u);
  const int idx  = ((lane + 16) & 31) << 2;
  return __int_as_float(__builtin_amdgcn_ds_bpermute(idx, __float_as_int(v)));
}

// ---------------------------------------------------------------------------
// Main pass. FINAL=false: zero-y-state run, record per-chunk finals p.
// FINAL=true: run with start states from scan, write output.
// One wave = 16 lockstep chunks; 16 samples/step via 5x WMMA f32 16x16x4.
//
// Assumed CDNA5 32-bit WMMA operand layouts (ISA 7.12.2):
//   A(16x4): lanes0-15 M=lane {K0,K1} in v0,v1; lanes16-31 M=lane-16 {K2,K3}
//   B(4x16): v0 = {K0 (lanes0-15), K2 (lanes16-31)}, v1 = {K1, K3}
//   C/D(16x16): vr = {M=r (lanes0-15), M=8+r (lanes16-31)}, N = lane%16
// ---------------------------------------------------------------------------
template<bool FINAL>
__global__ __launch_bounds__(128)
void pass_kernel(const float* __restrict__ x, float* __restrict__ out,
                 float* __restrict__ ws) {
  __shared__ __align__(16) float ldsIn [4][16*CH];
  __shared__ __align__(16) float ldsOut[4][16*CH];

  const int lane = (int)(threadIdx.x & 31u);
  const int wv   = (int)(threadIdx.x >> 5);
  const int gw   = blockIdx.x * 4 + wv;       // global wave id
  const int chunkBase = gw * 16;
  const int row    = chunkBase / CPR;         // 16 | CPR: wave never crosses rows
  const int cInRow = chunkBase % CPR;
  const int rowBase = row * TLEN;
  const int j    = lane & 15;                 // chunk within wave / matrix column
  const bool hi  = lane >= 16;
  const int koff = hi ? 2 : 0;

  // A operands: rows of T (and H) selected by lane per assumed layout
  const float* Tm = ws + WS_T;
  const float* Hm = ws + WS_H;
  const v2f a0v = *(const v2f*)(Tm + j*16 +  0 + koff);
  const v2f a1v = *(const v2f*)(Tm + j*16 +  4 + koff);
  const v2f a2v = *(const v2f*)(Tm + j*16 +  8 + koff);
  const v2f a3v = *(const v2f*)(Tm + j*16 + 12 + koff);
  const v2f asv = *(const v2f*)(Hm + j*4  + koff);

  // boundary-state B operand registers: lower lanes carry (y-1,y-2) of chunk j,
  // upper lanes carry (x-1,x-2) of chunk j.
  const int cstart = (cInRow + j) * CHUNK;    // in-row start position of chunk j
  float bs0, bs1;
  if (!hi) {
    if (FINAL) {
      const float* s = ws + WS_S;
      bs0 = s[2*(chunkBase + j) + 0];
      bs1 = s[2*(chunkBase + j) + 1];
    } else { bs0 = 0.f; bs1 = 0.f; }
  } else {
    bs0 = (cstart >= 1) ? x[rowBase + cstart - 1] : 0.f;
    bs1 = (cstart >= 2) ? x[rowBase + cstart - 2] : 0.f;
  }

  const int sChunk = lane >> 4;               // staging: chunk = 2*i + sChunk
  const int sOff   = (lane & 15) * 4;         // staging: sample offset in group

  for (int g = 0; g < GROUPS; ++g) {
    // ---- stage 16 chunks x 64 samples into LDS (256B-contiguous b128 runs) ----
    #pragma unroll
    for (int i = 0; i < 8; ++i) {
      const int ch = 2*i + sChunk;
      const float* gp = x + rowBase + (cInRow+ch)*CHUNK + g*64 + sOff;
      float* lp = &ldsIn[wv][ch*CH + sOff];
#if USE_ASYNC_LDS
      __builtin_amdgcn_global_load_async_to_lds_b128((v4i*)gp, (v4i*)lp, 0, 0);
#else
      *(float4*)lp = *(const float4*)gp;
#endif
    }
#if USE_ASYNC_LDS
    __builtin_amdgcn_s_wait_asynccnt(0);
#endif
    __syncthreads();

    // ---- 4 tiles of 16 samples ----
    #pragma unroll
    for (int tt = 0; tt < 4; ++tt) {
      const int base = tt * 16;
      const v2f b0v = *(const v2f*)&ldsIn[wv][j*CH + base +  0 + koff];
      const v2f b1v = *(const v2f*)&ldsIn[wv][j*CH + base +  4 + koff];
      const v2f b2v = *(const v2f*)&ldsIn[wv][j*CH + base +  8 + koff];
      const v2f b3v = *(const v2f*)&ldsIn[wv][j*CH + base + 12 + koff];
      v2f bsv; bsv[0] = bs0; bsv[1] = bs1;

      v8f acc = {};
      acc = __builtin_amdgcn_wmma_f32_16x16x4_f32(false, a0v, false, b0v, (short)0, acc, false, false);
      acc = __builtin_amdgcn_wmma_f32_16x16x4_f32(false, a1v, false, b1v, (short)0, acc, false, false);
      acc = __builtin_amdgcn_wmma_f32_16x16x4_f32(false, a2v, false, b2v, (short)0, acc, false, false);
      acc = __builtin_amdgcn_wmma_f32_16x16x4_f32(false, a3v, false, b3v, (short)0, acc, false, false);
      acc = __builtin_amdgcn_wmma_f32_16x16x4_f32(false, asv, false, bsv, (short)0, acc, false, false);

      // next boundary state: y rows 15,14 live on upper lanes of acc[7],acc[6];
      // x rows 15,14 live on upper lanes of b3v[1],b3v[0].
      const float ylast = swap_half(acc[7]);
      const float yprev = swap_half(acc[6]);
      bs0 = hi ? b3v[1] : ylast;
      bs1 = hi ? b3v[0] : yprev;

      if (FINAL) {
        float* o = &ldsOut[wv][j*CH + base + (hi ? 8 : 0)];
        *(float2*)(o+0) = make_float2(acc[0], acc[1]);
        *(float2*)(o+2) = make_float2(acc[2], acc[3]);
        *(float2*)(o+4) = make_float2(acc[4], acc[5]);
        *(float2*)(o+6) = make_float2(acc[6], acc[7]);
      }
    }
    __syncthreads();

    if (FINAL) {
      // ---- flush 16 chunks x 64 samples (256B-contiguous b128 runs) ----
      #pragma unroll
      for (int i = 0; i < 8; ++i) {
        const int ch = 2*i + sChunk;
        float* gp = out + rowBase + (cInRow+ch)*CHUNK + g*64 + sOff;
        float* lp = &ldsOut[wv][ch*CH + sOff];
#if USE_ASYNC_LDS
        __builtin_amdgcn_global_store_async_from_lds_b128((v4i*)gp, (v4i*)lp, 0, 0);
#else
        *(float4*)gp = *(const float4*)lp;
#endif
      }
#if USE_ASYNC_LDS
      __builtin_amdgcn_s_wait_asynccnt(0);   // LDS reads done before buffer reuse
#endif
      __syncthreads();
    }
  }

  if (!FINAL && !hi) {                        // record zero-state chunk finals
    float* p = ws + WS_P;
    p[2*(chunkBase + j) + 0] = bs0;           // y[L-1]
    p[2*(chunkBase + j) + 1] = bs1;           // y[L-2]
  }
}

// ---------------------------------------------------------------------------
extern "C" void kernel_launch(void* const* d_in, const int* in_sizes, int n_in,
                              void* d_out, int out_size, void* d_ws, size_t ws_size,
                              hipStream_t stream) {
  const float* x      = (const float*)d_in[0];
  const float* cutoff = (const float*)d_in[1];
  float* out = (float*)d_out;
  float* ws  = (float*)d_ws;   // needs ~267 KB

  setup_kernel<<<1, 1, 0, stream>>>(cutoff, ws);
  pass_kernel<false><<<NCHUNK/64, 128, 0, stream>>>(x, out, ws);
  scan_kernel<<<1, 64, 0, stream>>>(ws);
  pass_kernel<true ><<<NCHUNK/64, 128, 0, stream>>>(x, out, ws);
}